// StaticKVCache_91302414778672
// MI455X (gfx1250) — compile-verified
//
#include <hip/hip_runtime.h>

#ifndef __has_builtin
#define __has_builtin(x) 0
#endif

#if defined(__AMDGCN__) && \
    __has_builtin(__builtin_amdgcn_global_load_async_to_lds_b128) && \
    __has_builtin(__builtin_amdgcn_global_store_async_from_lds_b128)
#define USE_ASYNC_LDS 1
#else
#define USE_ASYNC_LDS 0
#endif

typedef float f4 __attribute__((ext_vector_type(4)));

namespace {
constexpr int  SINK   = 3072;          // sink rows
constexpr int  LOCALW = 12288;         // local ring rows
constexpr int  NEWL   = 1024;          // new rows
constexpr int  TOTAL  = SINK + LOCALW; // 15360 rows per tensor (K or V)
constexpr int  ROW4   = 512;           // float4 per (H*D) row = 16*128/4
constexpr long N4     = 2L * TOTAL * ROW4;  // 15,728,640 float4 total
constexpr int  TILE4  = 4 * 256;            // float4 per block tile (16 KB)
constexpr long NTILES = N4 / TILE4;         // 15,360 tiles, exact
}

// Map a flat output float4 index to its source float4 pointer.
__device__ __forceinline__ const f4* src_of(long idx, int wi, int nwi,
    const f4* __restrict__ sk, const f4* __restrict__ sv,
    const f4* __restrict__ lk, const f4* __restrict__ lv,
    const f4* __restrict__ nk, const f4* __restrict__ nv)
{
    int row = (int)(idx >> 9);        // 512 float4 per row
    int col = (int)(idx & 511);
    int kv  = row >= TOTAL;           // 0 = K, 1 = V
    int t   = row - (kv ? TOTAL : 0);
    const f4* s = kv ? sv : sk;
    const f4* l = kv ? lv : lk;
    const f4* n = kv ? nv : nk;
    if (t < SINK)
        return s + (long)t * ROW4 + col;
    int p = nwi + (t - SINK); if (p >= LOCALW) p -= LOCALW;   // ring position
    int d = p - wi;           if (d < 0)       d += LOCALW;   // offset into new range
    if (d < NEWL)
        return n + (long)d * ROW4 + col;
    return l + (long)p * ROW4 + col;
}

#if USE_ASYNC_LDS
// Builtin params are int4 pointers: global = AS(1) (prints as __device__),
// LDS = AS(3) (prints as __shared__/local).
typedef int v4i __attribute__((ext_vector_type(4)));
typedef __attribute__((address_space(1))) v4i g_v4i;
typedef __attribute__((address_space(3))) v4i l_v4i;

__device__ __forceinline__ g_v4i* as_global(const f4* p) {
    return (g_v4i*)(v4i*)p;
}
__device__ __forceinline__ l_v4i* as_lds(f4* p) {
    return (l_v4i*)(v4i*)p;
}

__device__ __forceinline__ void wait_async0() {
#if __has_builtin(__builtin_amdgcn_s_wait_asynccnt)
    __builtin_amdgcn_s_wait_asynccnt(0);
#else
    asm volatile("s_wait_asynccnt 0" ::: "memory");
#endif
}
#endif

__global__ __launch_bounds__(256) void kv_linearize(
    const f4* __restrict__ sk, const f4* __restrict__ sv,
    const f4* __restrict__ lk, const f4* __restrict__ lv,
    const f4* __restrict__ nk, const f4* __restrict__ nv,
    const int* __restrict__ widx, f4* __restrict__ out)
{
    int wi = widx[0] % LOCALW; if (wi < 0) wi += LOCALW;
    int nwi = wi + NEWL;       if (nwi >= LOCALW) nwi -= LOCALW;

#if USE_ASYNC_LDS
    // HBM -> LDS -> HBM copy on the async engines, 128-bit per lane,
    // double-buffered so a store batch and the next load batch overlap.
    __shared__ f4 smem[2][TILE4];     // 2 x 16 KB
    const int tid   = threadIdx.x;
    long tile       = blockIdx.x;
    const long step = gridDim.x;
    if (tile >= NTILES) return;

    int buf = 0;
    {   // prologue: loads for first tile
        long base = tile * TILE4;
#pragma unroll
        for (int u = 0; u < 4; ++u) {
            long idx = base + u * 256 + tid;
            const f4* s = src_of(idx, wi, nwi, sk, sv, lk, lv, nk, nv);
            __builtin_amdgcn_global_load_async_to_lds_b128(
                as_global(s), as_lds(&smem[buf][u * 256 + tid]), 0, 0);
        }
    }
    for (;;) {
        long next = tile + step;
        wait_async0();                         // this buf's loads (and prior stores) done
        {
            long base = tile * TILE4;
#pragma unroll
            for (int u = 0; u < 4; ++u) {
                long idx = base + u * 256 + tid;
                __builtin_amdgcn_global_store_async_from_lds_b128(
                    as_global(out + idx), as_lds(&smem[buf][u * 256 + tid]), 0, 0);
            }
        }
        if (next >= NTILES) break;
        buf ^= 1;                              // load next tile into other buffer,
        {                                      // overlapping the in-flight stores
            long base = next * TILE4;
#pragma unroll
            for (int u = 0; u < 4; ++u) {
                long idx = base + u * 256 + tid;
                const f4* s = src_of(idx, wi, nwi, sk, sv, lk, lv, nk, nv);
                __builtin_amdgcn_global_load_async_to_lds_b128(
                    as_global(s), as_lds(&smem[buf][u * 256 + tid]), 0, 0);
            }
        }
        tile = next;
    }
    wait_async0();                             // drain final stores
#else
    // Fallback: direct 128-bit grid-stride copy with nontemporal hints.
    long idx = (long)blockIdx.x * blockDim.x + threadIdx.x;
    const long stride = (long)gridDim.x * blockDim.x;
    for (; idx < N4; idx += stride) {
        const f4* s = src_of(idx, wi, nwi, sk, sv, lk, lv, nk, nv);
        __builtin_nontemporal_store(__builtin_nontemporal_load(s), out + idx);
    }
#endif
}

extern "C" void kernel_launch(void* const* d_in, const int* in_sizes, int n_in,
                              void* d_out, int out_size, void* d_ws, size_t ws_size,
                              hipStream_t stream) {
    (void)in_sizes; (void)n_in; (void)out_size; (void)d_ws; (void)ws_size;
    const f4*  sk   = (const f4*)d_in[0];
    const f4*  sv   = (const f4*)d_in[1];
    const f4*  lk   = (const f4*)d_in[2];
    const f4*  lv   = (const f4*)d_in[3];
    const f4*  nk   = (const f4*)d_in[4];
    const f4*  nv   = (const f4*)d_in[5];
    const int* widx = (const int*)d_in[6];
    f4* out = (f4*)d_out;

    // 1920 blocks x 256 threads: async path -> 8 tiles/block (pipelined);
    // fallback path -> ~32 float4 per thread grid-stride.
    kv_linearize<<<1920, 256, 0, stream>>>(sk, sv, lk, lv, nk, nv, widx, out);
}